// GCN_14491219657409
// MI455X (gfx1250) — compile-verified
//
#include <hip/hip_runtime.h>
#include <hip/hip_bf16.h>

#define NN 100000
#define NE 1600000

typedef __attribute__((ext_vector_type(2))) float v2f;
typedef __attribute__((ext_vector_type(8))) float v8f;

// ---------------------------------------------------------------------------
// Degree counting: scatter-add 1.0f per edge endpoint.
// ---------------------------------------------------------------------------
__global__ void __launch_bounds__(256)
k_count_deg(const int* __restrict__ src, const int* __restrict__ dst,
            float* __restrict__ outDeg, float* __restrict__ inDeg, int nE) {
  int e = blockIdx.x * 256 + threadIdx.x;
  if (e < nE) {
    atomicAdd(&outDeg[src[e]], 1.0f);
    atomicAdd(&inDeg[dst[e]], 1.0f);
  }
}

// In-place: deg -> rsqrt(max(deg, 1))
__global__ void __launch_bounds__(256)
k_coef(float* __restrict__ p, int n) {
  int i = blockIdx.x * 256 + threadIdx.x;
  if (i < n) p[i] = rsqrtf(fmaxf(p[i], 1.0f));
}

// ---------------------------------------------------------------------------
// FP32 WMMA GEMM:  Y[m,n] = rowScale[m] * sum_k X[m,k] * W[k,n]
//
// One wave per 16-row strip, covering ALL NT=C/16 column tiles, so the A
// fragment (2 VGPRs) is loaded once per K-step and reused by NT back-to-back
// V_WMMA_F32_16X16X4_F32 ops.  Accumulators: NT * v8f = up to 64 VGPRs.
// This cuts X traffic by NT x vs one-tile-per-wave (layer1: 820MB -> 102MB).
//
// V_WMMA_F32_16X16X4_F32 layouts (wave32):
//   A frag (16x4): lane -> {m = lane&15, kpair = (lane>>4)*2}, v2f
//   B frag (4x16): lane -> {n = lane&15, kpair = (lane>>4)*2}, v2f
//   C/D  (16x16):  lane -> {n = lane&15, half = lane>>4}, VGPR r -> row r+half*8
// EXEC must be all-ones: N = 6250*16 exactly, C%16==0, K%4==0, no divergence.
// ---------------------------------------------------------------------------
template <int NT>
__global__ void __launch_bounds__(32)
k_gemm_rowscale_wmma(const float* __restrict__ X, const float* __restrict__ W,
                     const float* __restrict__ rowScale, float* __restrict__ Y,
                     int K) {
  constexpr int C = NT * 16;
  const int lane  = threadIdx.x & 31;
  const int m0    = blockIdx.x << 4;
  const int idx15 = lane & 15;          // A: row in strip / B,D: col in tile
  const int koff  = (lane >> 4) << 1;   // 0 or 2

  const float* __restrict__ xrow  = X + (size_t)(m0 + idx15) * (size_t)K;
  const float* __restrict__ wbase = W + idx15;

  v8f acc[NT];
#pragma unroll
  for (int t = 0; t < NT; ++t) acc[t] = (v8f){};

#pragma unroll 2
  for (int k = 0; k < K; k += 4) {
    v2f a;
    a.x = xrow[k + koff];
    a.y = xrow[k + koff + 1];
    const float* __restrict__ wk = wbase + (size_t)(k + koff) * (size_t)C;
#pragma unroll
    for (int t = 0; t < NT; ++t) {
      v2f b;
      b.x = wk[t * 16];
      b.y = wk[t * 16 + C];
      acc[t] = __builtin_amdgcn_wmma_f32_16x16x4_f32(
          /*neg_a=*/false, a, /*neg_b=*/false, b,
          /*c_mod=*/(short)0, acc[t], /*reuse_a=*/false, /*reuse_b=*/false);
    }
  }

  const int half = lane >> 4;
#pragma unroll
  for (int r = 0; r < 8; ++r) {
    const int m   = m0 + r + half * 8;
    const float s = rowScale[m];
    float* __restrict__ yrow = Y + (size_t)m * (size_t)C + idx15;
#pragma unroll
    for (int t = 0; t < NT; ++t) {
      yrow[t * 16] = acc[t][r] * s;
    }
  }
}

// ---------------------------------------------------------------------------
// Edge gather + scatter-add.  For C=128 (gshift=5): one wave per edge, each
// lane moves a float4 (32 lanes * 16B = 512B coalesced gather; src/dst index
// loads are wave-uniform broadcasts).  For C=64 (gshift=4): half-wave/edge.
// y buffer (51.2 MB) is L2-resident, so gathers/atomics resolve in L2.
// ---------------------------------------------------------------------------
__global__ void __launch_bounds__(256)
k_edge_scatter(const float* __restrict__ Y, const int* __restrict__ src,
               const int* __restrict__ dst, float* __restrict__ M,
               int nE, int C, int gshift) {
  long tid = (long)blockIdx.x * 256 + threadIdx.x;
  if (tid >= ((long)nE << gshift)) return;
  const int e = (int)(tid >> gshift);
  const int g = ((int)tid) & ((1 << gshift) - 1);
  const int s = src[e];
  const int d = dst[e];
  const float4 v = reinterpret_cast<const float4*>(Y + (size_t)s * (size_t)C)[g];
  float* mp = M + (size_t)d * (size_t)C + (g << 2);
  atomicAdd(mp + 0, v.x);
  atomicAdd(mp + 1, v.y);
  atomicAdd(mp + 2, v.z);
  atomicAdd(mp + 3, v.w);
}

// ---------------------------------------------------------------------------
// Fused: h = relu(m * c_dst[row] + bias)  (stored to Hpre == h1/h2 output)
// then LayerNorm(h) * lnw + lnb (stored to Hln, feeds next layer).
// One wave32 per 128-channel row; 8 rows per 256-thread block (12500 blocks).
// ---------------------------------------------------------------------------
__global__ void __launch_bounds__(256)
k_finalize_ln128(const float* __restrict__ M, const float* __restrict__ cDst,
                 const float* __restrict__ bias, const float* __restrict__ lnw,
                 const float* __restrict__ lnb, float* __restrict__ Hpre,
                 float* __restrict__ Hln) {
  const int row  = blockIdx.x * 8 + (threadIdx.x >> 5);
  const int lane = threadIdx.x & 31;
  const float4 m4 = reinterpret_cast<const float4*>(M)[row * 32 + lane];
  const float4 b4 = reinterpret_cast<const float4*>(bias)[lane];
  const float cd  = cDst[row];

  const float h0 = fmaxf(fmaf(m4.x, cd, b4.x), 0.0f);
  const float h1 = fmaxf(fmaf(m4.y, cd, b4.y), 0.0f);
  const float h2 = fmaxf(fmaf(m4.z, cd, b4.z), 0.0f);
  const float h3 = fmaxf(fmaf(m4.w, cd, b4.w), 0.0f);
  reinterpret_cast<float4*>(Hpre)[row * 32 + lane] = make_float4(h0, h1, h2, h3);

  float s = h0 + h1 + h2 + h3;
  float q = h0 * h0 + h1 * h1 + h2 * h2 + h3 * h3;
#pragma unroll
  for (int off = 16; off > 0; off >>= 1) {
    s += __shfl_xor(s, off, 32);
    q += __shfl_xor(q, off, 32);
  }
  const float mean = s * (1.0f / 128.0f);
  const float var  = q * (1.0f / 128.0f) - mean * mean;
  const float inv  = rsqrtf(var + 1e-5f);

  const float4 w4  = reinterpret_cast<const float4*>(lnw)[lane];
  const float4 lb4 = reinterpret_cast<const float4*>(lnb)[lane];
  float4 o;
  o.x = (h0 - mean) * inv * w4.x + lb4.x;
  o.y = (h1 - mean) * inv * w4.y + lb4.y;
  o.z = (h2 - mean) * inv * w4.z + lb4.z;
  o.w = (h3 - mean) * inv * w4.w + lb4.w;
  reinterpret_cast<float4*>(Hln)[row * 32 + lane] = o;
}

// Layer 3 epilogue: out = m * c_dst[row] + bias  (C = 64, no relu, no LN).
__global__ void __launch_bounds__(256)
k_final64(const float* __restrict__ M, const float* __restrict__ cDst,
          const float* __restrict__ bias, float* __restrict__ Out) {
  const int i = blockIdx.x * 256 + threadIdx.x;  // N*64 = 6400000 exact
  const int row = i >> 6;
  const int c = i & 63;
  Out[i] = fmaf(M[i], cDst[row], bias[c]);
}

// ---------------------------------------------------------------------------
extern "C" void kernel_launch(void* const* d_in, const int* in_sizes, int n_in,
                              void* d_out, int out_size, void* d_ws, size_t ws_size,
                              hipStream_t stream) {
  const float* feats = (const float*)d_in[0];
  const int*   ei    = (const int*)d_in[1];
  const int*   src   = ei;
  const int*   dst   = ei + NE;
  const float* W1 = (const float*)d_in[2];  const float* b1 = (const float*)d_in[3];
  const float* W2 = (const float*)d_in[4];  const float* b2 = (const float*)d_in[5];
  const float* W3 = (const float*)d_in[6];  const float* b3 = (const float*)d_in[7];
  const float* ln1w = (const float*)d_in[8];  const float* ln1b = (const float*)d_in[9];
  const float* ln2w = (const float*)d_in[10]; const float* ln2b = (const float*)d_in[11];

  float* out  = (float*)d_out;
  float* h1o  = out;                       // [N,128]
  float* h2o  = out + (size_t)NN * 128;    // [N,128]
  float* o3   = h2o + (size_t)NN * 128;    // [N,64]

  float* ws    = (float*)d_ws;
  float* c_src = ws;                       // [N]
  float* c_dst = ws + NN;                  // [N]
  float* bufY  = ws + 2 * NN;              // [N,128] GEMM output (messages)
  float* bufM  = bufY + (size_t)NN * 128;  // [N,128] scatter accumulator
  float* bufH  = bufM + (size_t)NN * 128;  // [N,128] LN output (next input)

  // Degrees -> normalization coefficients
  hipMemsetAsync(c_src, 0, 2 * NN * sizeof(float), stream);
  k_count_deg<<<(NE + 255) / 256, 256, 0, stream>>>(src, dst, c_src, c_dst, NE);
  k_coef<<<(2 * NN + 255) / 256, 256, 0, stream>>>(c_src, 2 * NN);

  const int rowTiles = NN / 16;                      // 6250
  const int scat128  = (int)(((long)NE << 5) / 256); // 200000 blocks
  const int scat64   = (int)(((long)NE << 4) / 256); // 100000 blocks

  // ---- Layer 1: 256 -> 128
  k_gemm_rowscale_wmma<8><<<rowTiles, 32, 0, stream>>>(feats, W1, c_src, bufY, 256);
  hipMemsetAsync(bufM, 0, (size_t)NN * 128 * sizeof(float), stream);
  k_edge_scatter<<<scat128, 256, 0, stream>>>(bufY, src, dst, bufM, NE, 128, 5);
  k_finalize_ln128<<<NN / 8, 256, 0, stream>>>(bufM, c_dst, b1, ln1w, ln1b, h1o, bufH);

  // ---- Layer 2: 128 -> 128
  k_gemm_rowscale_wmma<8><<<rowTiles, 32, 0, stream>>>(bufH, W2, c_src, bufY, 128);
  hipMemsetAsync(bufM, 0, (size_t)NN * 128 * sizeof(float), stream);
  k_edge_scatter<<<scat128, 256, 0, stream>>>(bufY, src, dst, bufM, NE, 128, 5);
  k_finalize_ln128<<<NN / 8, 256, 0, stream>>>(bufM, c_dst, b2, ln2w, ln2b, h2o, bufH);

  // ---- Layer 3: 128 -> 64
  k_gemm_rowscale_wmma<4><<<rowTiles, 32, 0, stream>>>(bufH, W3, c_src, bufY, 128);
  hipMemsetAsync(bufM, 0, (size_t)NN * 64 * sizeof(float), stream);
  k_edge_scatter<<<scat64, 256, 0, stream>>>(bufY, src, dst, bufM, NE, 64, 4);
  k_final64<<<(NN * 64) / 256, 256, 0, stream>>>(bufM, c_dst, b3, o3);
}